// DistPool2D_31585189495157
// MI455X (gfx1250) — compile-verified
//
#include <hip/hip_runtime.h>

// DistPool2D (mode=MAX, 2x2/stride2, VALID, NHWC) on Gaussian moments.
// Memory: ~336 MB streamed once -> ~14.4 us floor at 23.3 TB/s.
// This revision attacks the VALU count (was 915/iter, i.e. ~23 us of VALU --
// compute-bound): raw v_rsq_f32/v_rcp_f32/v_exp_f32 TRANS ops instead of
// IEEE sqrtf/div/ocml-erff, and an Abramowitz-Stegun erf that *reuses* the
// pdf exponential (erf arg x=alpha*sqrt(1/2) => exp(-x^2)=exp(-alpha^2/2)).
// ~22 VALU + 3 TRANS per channel-merge, 12 merges/thread -> back under the
// HBM roof. Loads/stores stay b128 + TH_NT (stream exceeds 192MB L2).

typedef float v4f __attribute__((ext_vector_type(4)));

namespace {
constexpr int N = 32, H = 64, W = 64, C = 256;
constexpr int H2 = H / 2, W2 = W / 2, CV = C / 4;   // CV = vec4 channels
constexpr int TOTAL = N * H2 * W2 * CV;             // 2,097,152 vec4 sites
constexpr float kEps       = 1e-6f;
constexpr float kPdfNorm   = 0.3989422804014327f;   // 1/sqrt(2*pi)
constexpr float kSqrtHalf  = 0.7071067811865476f;   // sqrt(0.5)
constexpr float kNegHL2E   = -0.7213475204444817f;  // -0.5 * log2(e)
// Abramowitz & Stegun 7.1.26 erf constants (|err| <= 1.5e-7)
constexpr float kP  = 0.3275911f;
constexpr float kA1 = 0.254829592f;
constexpr float kA2 = -0.284496736f;
constexpr float kA3 = 1.421413741f;
constexpr float kA4 = -1.453152027f;
constexpr float kA5 = 1.061405429f;
}  // namespace

// Moment-matched Gaussian max of two Gaussians, scalar channel.
__device__ __forceinline__ void merge1(float m1, float v1, float m2, float v2,
                                       float& mo, float& vo) {
  float d     = v1 + v2 + kEps;                    // >= 1e-6, always normal
  float r     = __builtin_amdgcn_rsqf(d);          // v_rsq_f32
  float a     = d * r;                             // sqrt(d)
  float alpha = (m1 - m2) * r;                     // (m1-m2)/sqrt(d), no div
  // Shared exponential: E = exp(-alpha^2/2) = exp(-x^2), x = alpha*sqrt(1/2)
  float E     = __builtin_amdgcn_exp2f(alpha * alpha * kNegHL2E);  // v_exp_f32
  // erf(x) via A&S 7.1.26 on |x|, sign restored with bfi
  float x     = fabsf(alpha) * kSqrtHalf;
  float t     = __builtin_amdgcn_rcpf(fmaf(kP, x, 1.0f));          // v_rcp_f32
  float poly  = t * fmaf(t, fmaf(t, fmaf(t, fmaf(t, kA5, kA4), kA3), kA2), kA1);
  float erfa  = fmaf(-poly, E, 1.0f);
  float e     = copysignf(erfa, alpha);
  float cp    = fmaf(0.5f, e, 0.5f);               // 0.5*(1+erf)
  float cn    = fmaf(-0.5f, e, 0.5f);              // 0.5*(1-erf)
  float apdf  = a * (kPdfNorm * E);
  float mm    = fmaf(m1, cp, fmaf(m2, cn, apdf));
  float vv    = fmaf(fmaf(m1, m1, v1), cp,
                fmaf(fmaf(m2, m2, v2), cn,
                fmaf(m1 + m2, apdf, kEps - mm * mm)));
  mo = mm;
  vo = vv;
}

__device__ __forceinline__ void merge4(const v4f& m1, const v4f& v1,
                                       const v4f& m2, const v4f& v2,
                                       v4f& mo, v4f& vo) {
#pragma unroll
  for (int i = 0; i < 4; ++i) {
    float mm, vv;
    merge1(m1[i], v1[i], m2[i], v2[i], mm, vv);
    mo[i] = mm;
    vo[i] = vv;
  }
}

__global__ __launch_bounds__(256) void distpool2d_max_kernel(
    const float* __restrict__ xm, const float* __restrict__ xv,
    float* __restrict__ om, float* __restrict__ ov) {
  const int idx = blockIdx.x * 256 + threadIdx.x;
  if (idx >= TOTAL) return;

  // idx -> (n, ho, wo, c4); c4 fastest for coalescing along C.
  const int c4 = idx & (CV - 1);
  int t        = idx >> 6;        // / CV
  const int wo = t & (W2 - 1);
  t >>= 5;                        // / W2
  const int ho = t & (H2 - 1);
  const int n  = t >> 5;          // / H2

  // vec4 offset of input pixel (n, 2ho, 2wo, 4*c4)
  const int in_base = (((n * H + 2 * ho) * W) + 2 * wo) * CV + c4;
  const v4f* pm = (const v4f*)xm + in_base;
  const v4f* pv = (const v4f*)xv + in_base;

  // 8 independent b128 non-temporal loads issued as one clause (MLP).
  v4f m00 = __builtin_nontemporal_load(pm);
  v4f m01 = __builtin_nontemporal_load(pm + CV);            // w+1
  v4f m10 = __builtin_nontemporal_load(pm + W * CV);        // h+1
  v4f m11 = __builtin_nontemporal_load(pm + W * CV + CV);   // h+1, w+1
  v4f s00 = __builtin_nontemporal_load(pv);
  v4f s01 = __builtin_nontemporal_load(pv + CV);
  v4f s10 = __builtin_nontemporal_load(pv + W * CV);
  v4f s11 = __builtin_nontemporal_load(pv + W * CV + CV);

  // Same merge order as the reference: W pairs first, then H pair.
  v4f mA, vA, mB, vB, mO, vO;
  merge4(m00, s00, m01, s01, mA, vA);
  merge4(m10, s10, m11, s11, mB, vB);
  merge4(mA, vA, mB, vB, mO, vO);

  const int out_base = (((n * H2 + ho) * W2) + wo) * CV + c4;
  __builtin_nontemporal_store(mO, (v4f*)om + out_base);
  __builtin_nontemporal_store(vO, (v4f*)ov + out_base);
}

extern "C" void kernel_launch(void* const* d_in, const int* in_sizes, int n_in,
                              void* d_out, int out_size, void* d_ws, size_t ws_size,
                              hipStream_t stream) {
  const float* xm = (const float*)d_in[0];  // x_mean [32,64,64,256] f32
  const float* xv = (const float*)d_in[1];  // x_var  [32,64,64,256] f32
  float* om = (float*)d_out;                // m  [32,32,32,256] f32
  float* ov = om + (out_size / 2);          // v  [32,32,32,256] f32

  const int blocks = (TOTAL + 255) / 256;   // 8192 blocks x 256 threads (8 waves)
  distpool2d_max_kernel<<<blocks, 256, 0, stream>>>(xm, xv, om, ov);
}